// GaussianMomentDescriptor_77455440216255
// MI455X (gfx1250) — compile-verified
//
#include <hip/hip_runtime.h>
#include <cstdint>
#include <cstddef>

#define N_RADIAL 5
#define N_BASIS 7
#define N_SPECIES 119
#define N_ATOMS 20000
#define N_EDGES 600000
#define MOM_STRIDE 100   // per atom: m0[5] | m1[15] | m2c[5*6] | m3c[5*10]
#define OUT_STRIDE 460   // 5 + 15 + 15 + 15 + 35 + 125 + 125 + 125

typedef float v2f __attribute__((ext_vector_type(2)));
typedef float v8f __attribute__((ext_vector_type(8)));

// (i,j) -> compressed symmetric pair index (i<=j over dim 3)
__constant__ int I2TAB[9]  = {0,1,2, 1,3,4, 2,4,5};
// (i,j,k) -> compressed sorted-triple index (dim 3)
__constant__ int S3TAB[27] = {0,1,2,1,3,4,2,4,5,
                              1,3,4,3,6,7,4,7,8,
                              2,4,5,4,7,8,5,8,9};

// ---------------------------------------------------------------- zero ws
__global__ void zero_ws(float* __restrict__ ws, int n) {
  int i = blockIdx.x * blockDim.x + threadIdx.x;
  if (i < n) ws[i] = 0.0f;
}

// ---------------------------------------------------------------- edge stage
__global__ void edge_kernel(const float* __restrict__ dr_vec,
                            const int*   __restrict__ Z,
                            const int*   __restrict__ nbr,   // (2, N_EDGES)
                            const float* __restrict__ emb,   // (119,119,5,7)
                            float*       __restrict__ ws) {
  int e = blockIdx.x * blockDim.x + threadIdx.x;
  if (e >= N_EDGES) return;

  float dx = dr_vec[3*e+0], dy = dr_vec[3*e+1], dz = dr_vec[3*e+2];
  float dr  = sqrtf(dx*dx + dy*dy + dz*dz);
  float inv = 1.0f / (dr + 1e-5f);
  float nx = dx*inv, ny = dy*inv, nz = dz*inv;

  int ii = nbr[e];
  int jj = nbr[N_EDGES + e];
  int Zi = Z[ii], Zj = Z[jj];
  const float* cf = emb + ((size_t)Zj * N_SPECIES + (size_t)Zi) * (N_RADIAL * N_BASIS);

  // betta = 49/36, rad_norm = (2*betta/pi)^0.25, shifts = 0.5 + (5.5/7)*q
  const float betta    = 1.3611111111111112f;
  const float rad_norm = 0.96481287f;
  float b[7];
#pragma unroll
  for (int q = 0; q < 7; ++q) {
    float t = dr - (0.5f + 0.7857142857142857f * (float)q);
    b[q] = rad_norm * expf(-betta * t * t);
  }
  float drc = fminf(dr, 6.0f);
  float cut = 0.5f * (cosf(0.5235987755982988f * drc) + 1.0f);   // pi/6
  float scale = 0.3779644730092272f * cut;                       // 1/sqrt(7) * cutoff

  float rad[5];
#pragma unroll
  for (int r = 0; r < 5; ++r) {
    float s = 0.0f;
#pragma unroll
    for (int q = 0; q < 7; ++q) s += cf[r*7 + q] * b[q];
    rad[r] = s * scale;
  }

  // unique symmetric products of dn
  float p2[6]  = {nx*nx, nx*ny, nx*nz, ny*ny, ny*nz, nz*nz};
  float p3[10] = {p2[0]*nx, p2[0]*ny, p2[0]*nz, p2[1]*ny, p2[1]*nz,
                  p2[2]*nz, p2[3]*ny, p2[3]*nz, p2[4]*nz, p2[5]*nz};

  float* A = ws + (size_t)jj * MOM_STRIDE;   // segment by idx_j
#pragma unroll
  for (int r = 0; r < 5; ++r) {
    unsafeAtomicAdd(&A[r], rad[r]);
    unsafeAtomicAdd(&A[5 + r*3 + 0], rad[r]*nx);
    unsafeAtomicAdd(&A[5 + r*3 + 1], rad[r]*ny);
    unsafeAtomicAdd(&A[5 + r*3 + 2], rad[r]*nz);
#pragma unroll
    for (int q = 0; q < 6; ++q)
      unsafeAtomicAdd(&A[20 + r*6 + q], rad[r]*p2[q]);
#pragma unroll
    for (int q = 0; q < 10; ++q)
      unsafeAtomicAdd(&A[50 + r*10 + q], rad[r]*p3[q]);
  }
}

// -------------------------------------------------------- atom-stage helpers
// M3resh[(r,kk)][(i,j)] = M3[r][i][j][kk] from compressed m3c. row<16, col<12.
__device__ inline float m3resh(const float* Mp, int row, int col) {
  bool ok = (row < 15) & (col < 9);
  int rw = ok ? row : 0;
  int cl = ok ? col : 0;
  int r = rw / 3, kk = rw - 3*r;
  int i = cl / 3, j  = cl - 3*i;
  float v = Mp[50 + r*10 + S3TAB[(i*3 + j)*3 + kk]];
  return ok ? v : 0.0f;
}
// M2flat[s][(i,j)] from compressed m2c
__device__ inline float m2flatv(const float* Mp, int srow, int col) {
  bool ok = (srow < 5) & (col < 9);
  int sr = ok ? srow : 0;
  int cl = ok ? col : 0;
  float v = Mp[20 + sr*6 + I2TAB[cl]];
  return ok ? v : 0.0f;
}
// M2resh[(r,jj)][i] = M2[r][i][jj]
__device__ inline float m2resh(const float* Mp, int row, int col) {
  bool ok = (row < 15) & (col < 3);
  int rw = ok ? row : 0;
  int cl = ok ? col : 0;
  int r = rw / 3, jj = rw - 3*r;
  float v = Mp[20 + r*6 + I2TAB[cl*3 + jj]];
  return ok ? v : 0.0f;
}

__device__ inline void tri2dec(int q, int& r, int& s) {
  r = 0;
  while (q >= 5 - r) { q -= 5 - r; ++r; }
  s = r + q;
}
__device__ inline void tri3dec(int q, int& r, int& s, int& t) {
  r = 0;
  for (;;) { int c = (5 - r) * (6 - r) / 2; if (q < c) break; q -= c; ++r; }
  s = r;
  while (q >= 5 - s) { q -= 5 - s; ++s; }
  t = s + q;
}

// ---------------------------------------------------------------- atom stage
// One wave32 per atom; 8 atoms per 256-thread block. 20000 % 8 == 0.
__global__ __launch_bounds__(256) void atom_kernel(const float* __restrict__ ws,
                                                   float* __restrict__ out) {
  __shared__ float lds[8 * 896];
  int wave = threadIdx.x >> 5;
  int lane = threadIdx.x & 31;
  int atom = blockIdx.x * 8 + wave;

  const float* Mp = ws + (size_t)atom * MOM_STRIDE;
  float* base = lds + wave * 896;
  float* Tl  = base;          // 16x16 Gram of M3resh (15x9): T[(r,k),(s,l)]
  float* Xl  = base + 256;    // 16x16 Gram of M2resh (15x3): X[(r,j),(s,k)]
  float* Ul  = base + 512;    // 16x16: U[(r,k),s] = M3resh @ M2flat^T
  float* M2l = base + 768;    // 45: full M2[t][j][k]
  float* m1l = base + 816;    // 15
  float* m0l = base + 832;    // 5

  int mrow = lane & 15;            // A-row / B-col for this lane
  int koff = (lane >> 4) << 1;     // K pair held by this lane half

  // ---- WMMA stage (EXEC is all-ones here; operand build is branch-free) ----
  v8f T = {0,0,0,0,0,0,0,0};
  v8f U = {0,0,0,0,0,0,0,0};
#pragma unroll
  for (int s = 0; s < 3; ++s) {
    v2f a;  a.x  = m3resh (Mp, mrow, 4*s + koff);  a.y  = m3resh (Mp, mrow, 4*s + koff + 1);
    v2f bu; bu.x = m2flatv(Mp, mrow, 4*s + koff);  bu.y = m2flatv(Mp, mrow, 4*s + koff + 1);
    // Gram: B[k][n] = M3resh[n][4s+k]  ==> lane's B pair equals its A pair
    T = __builtin_amdgcn_wmma_f32_16x16x4_f32(false, a, false, a,  (short)0, T, false, false);
    U = __builtin_amdgcn_wmma_f32_16x16x4_f32(false, a, false, bu, (short)0, U, false, false);
  }
  v2f a2; a2.x = m2resh(Mp, mrow, koff); a2.y = m2resh(Mp, mrow, koff + 1);
  v8f X = {0,0,0,0,0,0,0,0};
  X = __builtin_amdgcn_wmma_f32_16x16x4_f32(false, a2, false, a2, (short)0, X, false, false);

  // C/D layout: lane L holds col N=L&15; VGPR v holds row M = v + 8*(L>>4)
#pragma unroll
  for (int v = 0; v < 8; ++v) {
    int Mr = v + ((lane >> 4) << 3);
    Tl[Mr*16 + mrow] = T[v];
    Xl[Mr*16 + mrow] = X[v];
    Ul[Mr*16 + mrow] = U[v];
  }

  // stage expanded moments into LDS
  for (int l = lane; l < 45; l += 32) {
    int t = l / 9, jk = l - 9*t;
    M2l[l] = Mp[20 + t*6 + I2TAB[jk]];
  }
  if (lane < 15) m1l[lane] = Mp[5 + lane];
  if (lane < 5)  m0l[lane] = Mp[lane];
  __syncthreads();

  // ---- emit 460 outputs per atom, 32 lanes cooperatively ----
  float* op = out + (size_t)atom * OUT_STRIDE;
  for (int p = lane; p < OUT_STRIDE; p += 32) {
    float val = 0.0f;
    if (p < 5) {                                  // c0
      val = m0l[p];
    } else if (p < 20) {                          // c1[r,s] = m1.m1
      int r, s; tri2dec(p - 5, r, s);
#pragma unroll
      for (int i = 0; i < 3; ++i) val += m1l[r*3+i] * m1l[s*3+i];
    } else if (p < 35) {                          // c2[r,s] = tr_j X
      int r, s; tri2dec(p - 20, r, s);
#pragma unroll
      for (int j = 0; j < 3; ++j) val += Xl[(3*r+j)*16 + (3*s+j)];
    } else if (p < 50) {                          // c3[r,s] = tr_k T
      int r, s; tri2dec(p - 35, r, s);
#pragma unroll
      for (int k = 0; k < 3; ++k) val += Tl[(3*r+k)*16 + (3*s+k)];
    } else if (p < 85) {                          // c4[r,s,t] = X : M2
      int r, s, t; tri3dec(p - 50, r, s, t);
#pragma unroll
      for (int j = 0; j < 3; ++j)
#pragma unroll
        for (int k = 0; k < 3; ++k)
          val += Xl[(3*r+j)*16 + (3*s+k)] * M2l[t*9 + j*3 + k];
    } else if (p < 210) {                         // c5[r,s,t] = m1 m1 : M2
      int q = p - 85;
      int r = q / 25, s = (q / 5) % 5, t = q % 5;
#pragma unroll
      for (int i = 0; i < 3; ++i)
#pragma unroll
        for (int j = 0; j < 3; ++j)
          val += m1l[r*3+i] * m1l[s*3+j] * M2l[t*9 + i*3 + j];
    } else if (p < 335) {                         // c6[r,s,t] = T : M2
      int q = p - 210;
      int r = q / 25, s = (q / 5) % 5, t = q % 5;
#pragma unroll
      for (int k = 0; k < 3; ++k)
#pragma unroll
        for (int l = 0; l < 3; ++l)
          val += Tl[(3*r+k)*16 + (3*s+l)] * M2l[t*9 + k*3 + l];
    } else {                                      // c7[r,s,t] = U . m1
      int q = p - 335;
      int r = q / 25, s = (q / 5) % 5, t = q % 5;
#pragma unroll
      for (int k = 0; k < 3; ++k)
        val += Ul[(3*r+k)*16 + s] * m1l[t*3 + k];
    }
    op[p] = val;
  }
}

// ---------------------------------------------------------------- launcher
extern "C" void kernel_launch(void* const* d_in, const int* in_sizes, int n_in,
                              void* d_out, int out_size, void* d_ws, size_t ws_size,
                              hipStream_t stream) {
  const float* dr_vec = (const float*)d_in[0];
  const int*   Z      = (const int*)  d_in[1];
  const int*   nbr    = (const int*)  d_in[2];
  const float* emb    = (const float*)d_in[3];
  float* out = (float*)d_out;
  float* ws  = (float*)d_ws;

  const int nws = N_ATOMS * MOM_STRIDE;     // 2,000,000 floats = 8 MB
  zero_ws  <<<(nws + 255) / 256,       256, 0, stream>>>(ws, nws);
  edge_kernel<<<(N_EDGES + 255) / 256, 256, 0, stream>>>(dr_vec, Z, nbr, emb, ws);
  atom_kernel<<<N_ATOMS / 8,           256, 0, stream>>>(ws, out);
}